// SMPLDeformer_24902220382388
// MI455X (gfx1250) — compile-verified
//
#include <hip/hip_runtime.h>

#define NV      6890
#define NJ      24
#define KNN     7
#define CHUNK_T 216                 // 16-vertex tiles per LDS chunk
#define CHUNK_V (CHUNK_T * 16)      // 3456 vertices per chunk
#define NCHUNK  2                   // covers 6912 >= 6890 (padded with sentinel)

typedef __attribute__((ext_vector_type(2))) float v2f;
typedef __attribute__((ext_vector_type(8))) float v8f;

// Gate-then-insert: one fminf-tree gate per 8 candidates; branchless bubble when taken.
__device__ __attribute__((always_inline)) inline void topk_insert8(
    const v8f& d, int base, float (&best)[KNN], int (&bidx)[KNN]) {
  float mA = fminf(fminf(d[0], d[1]), fminf(d[2], d[3]));
  float mB = fminf(fminf(d[4], d[5]), fminf(d[6], d[7]));
  if (fminf(mA, mB) < best[KNN - 1]) {
    #pragma unroll
    for (int r = 0; r < 8; ++r) {
      float dv = d[r];
      if (dv < best[KNN - 1]) {
        best[KNN - 1] = dv; bidx[KNN - 1] = base + r;
        #pragma unroll
        for (int k = KNN - 1; k > 0; --k) {   // branchless bubble-up (cndmask)
          bool  sw = best[k] < best[k - 1];
          float bl = sw ? best[k]     : best[k - 1];
          float bh = sw ? best[k - 1] : best[k];
          int   il = sw ? bidx[k]     : bidx[k - 1];
          int   ih = sw ? bidx[k - 1] : bidx[k];
          best[k - 1] = bl; best[k] = bh;
          bidx[k - 1] = il; bidx[k] = ih;
        }
      }
    }
  }
}

__global__ __launch_bounds__(256) void smpl_knn_ilbs(
    const float* __restrict__ xq,     // [N,3]
    const float* __restrict__ tfs,    // [24,4,4]
    const float* __restrict__ verts,  // [V,3]
    const float* __restrict__ wtab,   // [V,24]
    float* __restrict__ out,          // [3N] x_out  ++ [N] mask
    int nq)
{
  __shared__ float ldsV[CHUNK_V * 4 + 128]; // A-operand verts + prefetch pad (~55.8 KB)
  __shared__ float ldsT[NJ * 12];           // top 3 rows of each bone transform

  const int tid    = threadIdx.x;
  const int lane   = tid & 31;
  const int wave   = tid >> 5;
  const int laneLo = lane & 15;
  const int hi     = lane >> 4;             // 0: K=0..1 / verts 0..7, 1: K=2..3 / verts 8..15

  // stage bone transforms once (rows 0..2; row 3 is [0,0,0,1] by construction)
  for (int s = tid; s < NJ * 12; s += 256) {
    int j = s / 12, e = s - j * 12;
    ldsT[s] = tfs[j * 16 + e];
  }

  const int  qbase  = (blockIdx.x * 8 + wave) * 16;
  const bool active = (qbase < nq);

  // B operand: B[k][n] = (qx,qy,qz,1)^T for query n. Lane n holds K=0,1; lane n+16 K=2,3.
  float qx = 0.f, qy = 0.f, qz = 0.f, pp = 0.f;
  v2f bmat; bmat.x = 0.f; bmat.y = 0.f;
  if (active) {
    int ql = qbase + laneLo;
    if (ql >= nq) ql = nq - 1;              // clamp (keeps EXEC full for WMMA)
    qx = xq[3 * ql]; qy = xq[3 * ql + 1]; qz = xq[3 * ql + 2];
    pp = qx * qx + qy * qy + qz * qz;
    bmat.x = hi ? qz : qx;
    bmat.y = hi ? 1.0f : qy;
  }
  v8f cmat;                                 // C = ||p||^2 so D is the exact squared distance
  #pragma unroll
  for (int r = 0; r < 8; ++r) cmat[r] = pp;

  float best[KNN]; int bidx[KNN];           // sorted ascending, lane-local top-7
  #pragma unroll
  for (int k = 0; k < KNN; ++k) { best[k] = 3.0e30f; bidx[k] = 0; }

  const float* aPtr = &ldsV[laneLo * 4 + hi * 2];

  for (int ch = 0; ch < NCHUNK; ++ch) {
    __syncthreads();
    const int vb = ch * CHUNK_V;
    // stage this chunk of vertices in A-operand form; pad with huge |v|^2 sentinel
    for (int s = tid; s < CHUNK_V; s += 256) {
      int v = vb + s;
      float vx = 0.f, vy = 0.f, vz = 0.f, w = 1.0e30f;
      if (v < NV) {
        vx = verts[3 * v]; vy = verts[3 * v + 1]; vz = verts[3 * v + 2];
        w  = vx * vx + vy * vy + vz * vz;
        vx *= -2.0f; vy *= -2.0f; vz *= -2.0f;
      }
      ldsV[4 * s + 0] = vx; ldsV[4 * s + 1] = vy;
      ldsV[4 * s + 2] = vz; ldsV[4 * s + 3] = w;
    }
    __syncthreads();

    if (active) {                           // wave-uniform -> EXEC all ones at WMMA
      const float* ap = aPtr;
      v2f a0 = *(const v2f*)(ap);           // prime the 2-tile pipeline
      v2f a1 = *(const v2f*)(ap + 64);
      for (int t = 0; t < CHUNK_T; t += 2) {
        v2f n0 = *(const v2f*)(ap + 128);   // prefetch next pair (pad keeps in-bounds)
        v2f n1 = *(const v2f*)(ap + 192);
        v8f d0 = __builtin_amdgcn_wmma_f32_16x16x4_f32(
            false, a0, false, bmat, (short)0, cmat, false, false);
        topk_insert8(d0, vb + t * 16 + hi * 8, best, bidx);
        v8f d1 = __builtin_amdgcn_wmma_f32_16x16x4_f32(
            false, a1, false, bmat, (short)0, cmat, false, false);
        topk_insert8(d1, vb + (t + 1) * 16 + hi * 8, best, bidx);
        a0 = n0; a1 = n1; ap += 128;
      }
    }
  }

  if (!active) return;

  // merge lane n (verts m%16<8) with lane n+16 (verts m%16>=8): both lists sorted asc.
  float pd[KNN]; int pi[KNN];
  #pragma unroll
  for (int k = 0; k < KNN; ++k) {
    pd[k] = __shfl_xor(best[k], 16, 32);
    pi[k] = __shfl_xor(bidx[k], 16, 32);
  }
  // bitonic lower-half merge: md[k] = min(A[k], B[6-k]) is the 7 smallest (bitonic order)
  float md[8]; int mi[8];
  #pragma unroll
  for (int k = 0; k < KNN; ++k) {
    bool ta = best[k] <= pd[KNN - 1 - k];
    md[k] = ta ? best[k] : pd[KNN - 1 - k];
    mi[k] = ta ? bidx[k] : pi[KNN - 1 - k];
  }
  md[7] = 3.1e30f; mi[7] = 0;
  #define CEX(A2, B2) { bool sw = md[B2] < md[A2];                         \
                        float tl = sw ? md[B2] : md[A2];                   \
                        float th = sw ? md[A2] : md[B2];                   \
                        int   jl = sw ? mi[B2] : mi[A2];                   \
                        int   jh = sw ? mi[A2] : mi[B2];                   \
                        md[A2] = tl; md[B2] = th; mi[A2] = jl; mi[B2] = jh; }
  CEX(0,4) CEX(1,5) CEX(2,6) CEX(3,7)
  CEX(0,2) CEX(1,3) CEX(4,6) CEX(5,7)
  CEX(0,1) CEX(2,3) CEX(4,5) CEX(6,7)
  #undef CEX

  if (hi != 0) return;                      // one lane per query finalizes
  const int q = qbase + laneLo;
  if (q >= nq) return;

  // conf = exp(-min(max(d2,0),4)), normalized
  float conf[KNN]; float csum = 0.f;
  #pragma unroll
  for (int k = 0; k < KNN; ++k) {
    float d2 = fminf(fmaxf(md[k], 0.f), 4.0f);
    conf[k] = __expf(-d2);
    csum += conf[k];
  }
  const float cs = 1.0f / csum;

  // weights[24] = sum_k conf_k * wtab[idx_k][:]   (row stride 96B -> float4 aligned)
  float wj[NJ];
  #pragma unroll
  for (int j = 0; j < NJ; ++j) wj[j] = 0.f;
  #pragma unroll
  for (int k = 0; k < KNN; ++k) {
    const float4* row = (const float4*)(wtab + (size_t)mi[k] * NJ);
    float ck = conf[k] * cs;
    #pragma unroll
    for (int c = 0; c < 6; ++c) {
      float4 r4 = row[c];
      wj[4*c+0] += ck * r4.x; wj[4*c+1] += ck * r4.y;
      wj[4*c+2] += ck * r4.z; wj[4*c+3] += ck * r4.w;
    }
  }

  // blended affine transform (top 3 rows), then closed-form affine inverse applied to q
  float M[12];
  #pragma unroll
  for (int e = 0; e < 12; ++e) M[e] = 0.f;
  #pragma unroll
  for (int j = 0; j < NJ; ++j) {
    float w = wj[j];
    #pragma unroll
    for (int e = 0; e < 12; ++e) M[e] += w * ldsT[j * 12 + e];
  }
  float m0=M[0], m1=M[1], m2=M[2],  m3=M[3];
  float m4=M[4], m5=M[5], m6=M[6],  m7=M[7];
  float m8=M[8], m9=M[9], m10=M[10], m11=M[11];
  float c00 = m5*m10 - m6*m9;
  float c01 = m6*m8  - m4*m10;
  float c02 = m4*m9  - m5*m8;
  float det = m0*c00 + m1*c01 + m2*c02;
  float rdet = 1.0f / det;
  float a01 = m2*m9  - m1*m10, a02 = m1*m6 - m2*m5;
  float a11 = m0*m10 - m2*m8,  a12 = m2*m4 - m0*m6;
  float a21 = m1*m8  - m0*m9,  a22 = m0*m5 - m1*m4;
  float px = qx - m3, py = qy - m7, pz = qz - m11;

  out[3*q + 0] = (c00*px + a01*py + a02*pz) * rdet;
  out[3*q + 1] = (c01*px + a11*py + a12*pz) * rdet;
  out[3*q + 2] = (c02*px + a21*py + a22*pz) * rdet;
  out[3*nq + q] = (fmaxf(md[0], 0.f) > 0.01f) ? 1.0f : 0.0f;  // sqrt(d2)>0.1
}

extern "C" void kernel_launch(void* const* d_in, const int* in_sizes, int n_in,
                              void* d_out, int out_size, void* d_ws, size_t ws_size,
                              hipStream_t stream) {
  const float* x     = (const float*)d_in[0];  // [N,3]
  const float* tfs   = (const float*)d_in[1];  // [1,24,4,4]
  const float* verts = (const float*)d_in[2];  // [1,V,3]
  const float* wtab  = (const float*)d_in[3];  // [1,V,24]
  float* out = (float*)d_out;
  const int nq = in_sizes[0] / 3;
  const int nwaves  = (nq + 15) / 16;          // 16 queries per wave
  const int nblocks = (nwaves + 7) / 8;        // 8 waves per block
  hipLaunchKernelGGL(smpl_knn_ilbs, dim3(nblocks), dim3(256), 0, stream,
                     x, tfs, verts, wtab, out, nq);
}